// RelativeBucketedTimeAndPositionBasedBias_81664508166288
// MI455X (gfx1250) — compile-verified
//
#include <hip/hip_runtime.h>
#include <stdint.h>

#define NTHREADS 512
#define RPB 8          // rows per block
#define TILE_N 2048    // fast path requires N == 4*NTHREADS == 2048

// ---------------------------------------------------------------------------
// CDNA5 async global->LDS copies (ASYNCcnt-tracked), CDNA5 ISA.
// GVS mode: mem_addr = SADDR(64) + VADDR(32). VDST VGPR holds LDS byte address.
// ---------------------------------------------------------------------------
__device__ __forceinline__ void async_load_b32_to_lds(void* lds_ptr,
                                                      const void* gbase,
                                                      uint32_t byte_off) {
    uint32_t laddr = (uint32_t)(uintptr_t)lds_ptr;
    uint64_t sbase = (uint64_t)(uintptr_t)gbase;
    asm volatile("global_load_async_to_lds_b32 %0, %1, %2"
                 :: "v"(laddr), "v"(byte_off), "s"(sbase)
                 : "memory");
}

__device__ __forceinline__ void async_load_b128_to_lds(void* lds_ptr,
                                                       const void* gbase,
                                                       uint32_t byte_off) {
    uint32_t laddr = (uint32_t)(uintptr_t)lds_ptr;
    uint64_t sbase = (uint64_t)(uintptr_t)gbase;
    asm volatile("global_load_async_to_lds_b128 %0, %1, %2"
                 :: "v"(laddr), "v"(byte_off), "s"(sbase)
                 : "memory");
}

__device__ __forceinline__ void wait_async_all() {
#if defined(__has_builtin)
#if __has_builtin(__builtin_amdgcn_s_wait_asynccnt)
    __builtin_amdgcn_s_wait_asynccnt(0);
    return;
#endif
#endif
    asm volatile("s_wait_asynccnt 0" ::: "memory");
}

// log(x)/0.301 == log2(x) * (ln2/0.301); v_log_f32 is native log2
#define LOG_SCALE ((float)(0.6931471805599453 / 0.301))

__device__ __forceinline__ float bias_elem(float tn, float tj, float p,
                                           const float* s_tsw, int nbk) {
    float d = fabsf(tn - tj);
    d = fminf(fmaxf(d, 1.0f), 1.0e9f);
    int bk = (int)(__log2f(d) * LOG_SCALE);     // d >= 1 -> result >= 0, trunc == floor
    bk = max(0, min(bk, nbk));
    return p + s_tsw[bk];
}

// ---------------------------------------------------------------------------
// Fast path: blockIdx.x = row strip (8 rows), blockIdx.y = batch. N fixed 2048.
// ---------------------------------------------------------------------------
__global__ __launch_bounds__(NTHREADS)
void bias_tile_kernel(const int* __restrict__ ts,
                      const float* __restrict__ pos,
                      const float* __restrict__ tsw,
                      float* __restrict__ out,
                      int NT) {
    constexpr int N = TILE_N;
    __shared__ __align__(16) int   s_ti[N];            // raw i32 timestamp row
    __shared__ __align__(16) float s_pos[N + RPB];     // 2055 used + pad
    __shared__ __align__(16) float s_tsw[136];         // NT (129) used

    const int tid = threadIdx.x;
    const int b   = blockIdx.y;
    const int i0  = blockIdx.x * RPB;
    const int posBase = (N - 1) - (i0 + RPB - 1);      // multiple of 8 -> 16B-aligned
    const int c0  = tid * 4;

    // ---- async stage (one b128 per thread per array, tails as b32) ----
    async_load_b128_to_lds(&s_ti[c0], ts, (uint32_t)((b * N + c0) * 4));
    async_load_b128_to_lds(&s_pos[c0], pos, (uint32_t)((posBase + c0) * 4));
    if (tid < RPB - 1)                                 // pos tail: indices N..N+6, global <= 4094-i0
        async_load_b32_to_lds(&s_pos[N + tid], pos, (uint32_t)((posBase + N + tid) * 4));
    if (c0 + 3 < NT)                                   // ts_w bulk: 32 threads x b128
        async_load_b128_to_lds(&s_tsw[c0], tsw, (uint32_t)(c0 * 4));
    {
        const int rem = NT & 3;                        // ts_w tail (element 128)
        if (tid < rem)
            async_load_b32_to_lds(&s_tsw[(NT & ~3) + tid], tsw,
                                  (uint32_t)(((NT & ~3) + tid) * 4));
    }

    wait_async_all();
    __syncthreads();

    // ---- this thread owns columns c0..c0+3 for all 8 rows ----
    const int4 tiv = *(const int4*)&s_ti[c0];
    const float tjx = (float)tiv.x, tjy = (float)tiv.y,
                tjz = (float)tiv.z, tjw = (float)tiv.w;

    float pr[11];                                      // pos window this thread needs
    {
        const float4 p0 = *(const float4*)&s_pos[c0];
        const float4 p1 = *(const float4*)&s_pos[c0 + 4];
        pr[0] = p0.x; pr[1] = p0.y; pr[2] = p0.z; pr[3] = p0.w;
        pr[4] = p1.x; pr[5] = p1.y; pr[6] = p1.z; pr[7] = p1.w;
        pr[8] = s_pos[c0 + 8]; pr[9] = s_pos[c0 + 9]; pr[10] = s_pos[c0 + 10];
    }

    const int nbk = NT - 1;
#pragma unroll
    for (int di = 0; di < RPB; ++di) {
        const int i  = i0 + di;
        const int sh = (RPB - 1) - di;                 // pos[j-i+N-1] = s_pos[j+sh]
        const float tn = (float)s_ti[min(i + 1, N - 1)];
        float4 r;
        r.x = bias_elem(tn, tjx, pr[sh + 0], s_tsw, nbk);
        r.y = bias_elem(tn, tjy, pr[sh + 1], s_tsw, nbk);
        r.z = bias_elem(tn, tjz, pr[sh + 2], s_tsw, nbk);
        r.w = bias_elem(tn, tjw, pr[sh + 3], s_tsw, nbk);
        *(float4*)(out + ((size_t)(b * N + i)) * N + c0) = r;
    }
}

// ---------------------------------------------------------------------------
// Generic fallback (any shape), grid-stride elementwise.
// ---------------------------------------------------------------------------
__global__ void bias_generic_kernel(const int* __restrict__ ts,
                                    const float* __restrict__ pos,
                                    const float* __restrict__ tsw,
                                    float* __restrict__ out,
                                    int N, int B, int NT) {
    const size_t total = (size_t)B * N * N;
    const int nbk = NT - 1;
    for (size_t idx = (size_t)blockIdx.x * blockDim.x + threadIdx.x; idx < total;
         idx += (size_t)gridDim.x * blockDim.x) {
        const int j = (int)(idx % N);
        const size_t rest = idx / N;
        const int i = (int)(rest % N);
        const int b = (int)(rest / N);
        const float tn = (float)ts[b * N + min(i + 1, N - 1)];
        const float tjv = (float)ts[b * N + j];
        float d = fminf(fmaxf(fabsf(tn - tjv), 1.0f), 1.0e9f);
        int bk = (int)(__log2f(d) * LOG_SCALE);
        bk = max(0, min(bk, nbk));
        out[idx] = pos[j - i + N - 1] + tsw[bk];
    }
}

extern "C" void kernel_launch(void* const* d_in, const int* in_sizes, int n_in,
                              void* d_out, int out_size, void* d_ws, size_t ws_size,
                              hipStream_t stream) {
    const int*   ts  = (const int*)d_in[0];     // (B, N) int32
    const float* pos = (const float*)d_in[1];   // (2N-1,) f32
    const float* tsw = (const float*)d_in[2];   // (NB+1,) f32
    float*       out = (float*)d_out;           // (B, N, N) f32

    const int N  = (in_sizes[1] + 1) / 2;
    const int B  = in_sizes[0] / N;
    const int NT = in_sizes[2];

    if (N == TILE_N && NT <= 132) {
        dim3 grid(N / RPB, B);
        bias_tile_kernel<<<grid, NTHREADS, 0, stream>>>(ts, pos, tsw, out, NT);
    } else {
        size_t total = (size_t)B * N * N;
        size_t want = (total + 255) / 256;
        int blocks = (int)((want < (size_t)65535) ? want : (size_t)65535);
        bias_generic_kernel<<<blocks, 256, 0, stream>>>(ts, pos, tsw, out, N, B, NT);
    }
}